// VectorQuantize_34694745817195
// MI455X (gfx1250) — compile-verified
//
#include <hip/hip_runtime.h>
#include <hip/hip_bf16.h>

typedef float v2f __attribute__((ext_vector_type(2)));
typedef float v8f __attribute__((ext_vector_type(8)));

#define B_      16
#define DIN     1024
#define T_      4096
#define DC      8
#define K_      1024
#define TILE_T  256
#define NT_TILES (T_ / TILE_T)   // 16

// ---------------- output offsets (floats) ----------------
#define OUT_ZQ     0
#define OUT_COMMIT 67108864
#define OUT_CBL    67108880
#define OUT_IDX    67108896
#define OUT_ZE     67174432

// ---------------- workspace offsets (floats) -------------
#define WS_CN     0          // 1024*8 normalized codebook
#define WS_CN2    8192       // 1024   ||cn_k||^2
#define WS_ZQROT  9216       // B*T*8  rotated/scaled zq
#define WS_PART   533504     // B*NT_TILES loss partials

// =========================================================
// Kernel 0: normalize codebook, precompute ||cn||^2
// =========================================================
__global__ __launch_bounds__(256) void vq_prep(const float* __restrict__ cb,
                                               float* __restrict__ cn,
                                               float* __restrict__ cn2) {
  int k = blockIdx.x * 256 + threadIdx.x;
  if (k < K_) {
    float v[8]; float s = 0.f;
#pragma unroll
    for (int d = 0; d < 8; ++d) { v[d] = cb[k * 8 + d]; s = fmaf(v[d], v[d], s); }
    float inv = 1.0f / fmaxf(sqrtf(s), 1e-12f);
    float s2 = 0.f;
#pragma unroll
    for (int d = 0; d < 8; ++d) {
      float nv = v[d] * inv;
      cn[k * 8 + d] = nv;
      s2 = fmaf(nv, nv, s2);
    }
    cn2[k] = s2;
  }
}

// =========================================================
// Kernel 1: fused in_proj -> z_e, WMMA cosine argmax,
//           codebook gather, losses, rotation trick
// grid: (T/256, B), block: 256 (8 waves)
// =========================================================
__global__ __launch_bounds__(256) void vq_main(
    const float* __restrict__ z,     // [B, DIN, T]
    const float* __restrict__ in_w,  // [DC, DIN]
    const float* __restrict__ in_b,  // [DC]
    const float* __restrict__ cb,    // [K, DC] raw codebook
    const float* __restrict__ cn,    // [K, DC] normalized
    const float* __restrict__ cn2,   // [K]
    float* __restrict__ ze_out,      // [B, DC, T]
    int*   __restrict__ idx_out,     // [B, T]
    float* __restrict__ zqrot,       // [B, T, DC]
    float* __restrict__ part)        // [B, NT_TILES]
{
  __shared__ float s_w[DIN * DC];       // 32KB: in_w^T, then cn
  __shared__ float s_zt[TILE_T * DC];   // 8KB raw z_e rows
  __shared__ float s_ie[TILE_T];        // 1/max(||zt||,1e-12)
  __shared__ float s_cn2[K_];           // 4KB
  __shared__ int   s_idx[TILE_T];
  __shared__ float s_red[256];

  const int tid = threadIdx.x;
  const int b   = blockIdx.y;
  const int t0  = blockIdx.x * TILE_T;

  // ---- stage in_w transposed: s_w[d*8+o] = in_w[o*DIN+d] ----
  for (int i = tid; i < DIN * DC; i += 256) {
    int d = i >> 3, o = i & 7;
    s_w[i] = in_w[o * DIN + d];
  }
  __syncthreads();

  // ---- in_proj: one timestep per thread (coalesced over t) ----
  float acc[8];
#pragma unroll
  for (int o = 0; o < 8; ++o) acc[o] = in_b[o];
  const float* zp = z + ((size_t)b * DIN) * T_ + (t0 + tid);
#pragma unroll 4
  for (int d = 0; d < DIN; ++d) {
    float zv = zp[(size_t)d * T_];
    const float* wr = &s_w[d * 8];         // same addr all lanes -> LDS broadcast
#pragma unroll
    for (int o = 0; o < 8; ++o) acc[o] = fmaf(wr[o], zv, acc[o]);
  }

  // ---- write z_e; stash row + inverse norm in LDS ----
  {
    float* zep = ze_out + ((size_t)b * DC) * T_ + (t0 + tid);
    float n2 = 0.f;
#pragma unroll
    for (int o = 0; o < 8; ++o) {
      zep[(size_t)o * T_] = acc[o];
      s_zt[tid * 8 + o] = acc[o];
      n2 = fmaf(acc[o], acc[o], n2);
    }
    s_ie[tid] = 1.0f / fmaxf(sqrtf(n2), 1e-12f);
  }
  __syncthreads();

  // ---- reload LDS with normalized codebook ----
  for (int i = tid; i < K_ * DC; i += 256) s_w[i] = cn[i];
  for (int i = tid; i < K_; i += 256) s_cn2[i] = cn2[i];
  __syncthreads();

  // ---- WMMA scoring: score = en . cn_k - 0.5*||cn_k||^2 ----
  const int lane = tid & 31;
  const int w    = tid >> 5;
  const int r    = lane & 15;   // A: row M / B: col N
  const int hi   = lane >> 4;   // K-pair selector

#pragma unroll
  for (int rt = 0; rt < 2; ++rt) {
    const int rowbase = w * 32 + rt * 16;
    const float  sa = s_ie[rowbase + r];           // scale row to e_norm
    const float* ar = &s_zt[(rowbase + r) * 8 + hi * 2];
    v2f a0; a0.x = ar[0] * sa; a0.y = ar[1] * sa;  // K = 2hi, 2hi+1
    v2f a1; a1.x = ar[4] * sa; a1.y = ar[5] * sa;  // K = 4+2hi, 5+2hi

    float best[8]; int bidx[8];
#pragma unroll
    for (int j = 0; j < 8; ++j) { best[j] = -3.4e38f; bidx[j] = 0; }

    for (int kt = 0; kt < K_ / 16; ++kt) {
      const float* br = &s_w[(kt * 16 + r) * 8 + hi * 2];
      v2f b0; b0.x = br[0]; b0.y = br[1];
      v2f b1; b1.x = br[4]; b1.y = br[5];
      v8f c = {};
      c = __builtin_amdgcn_wmma_f32_16x16x4_f32(false, a0, false, b0,
                                                (short)0, c, false, false);
      c = __builtin_amdgcn_wmma_f32_16x16x4_f32(false, a1, false, b1,
                                                (short)0, c, false, false);
      const int   kcol = kt * 16 + r;
      const float adj  = 0.5f * s_cn2[kcol];
#pragma unroll
      for (int j = 0; j < 8; ++j) {
        float v = c[j] - adj;
        if (v > best[j]) { best[j] = v; bidx[j] = kcol; }
      }
    }
    // butterfly argmax over the 16 N-lanes of each half (first-index ties)
#pragma unroll
    for (int off = 8; off >= 1; off >>= 1) {
#pragma unroll
      for (int j = 0; j < 8; ++j) {
        float ob = __shfl_xor(best[j], off, 32);
        int   oi = __shfl_xor(bidx[j], off, 32);
        if (ob > best[j] || (ob == best[j] && oi < bidx[j])) {
          best[j] = ob; bidx[j] = oi;
        }
      }
    }
    if (r == 0) {
#pragma unroll
      for (int j = 0; j < 8; ++j) s_idx[rowbase + hi * 8 + j] = bidx[j];
    }
  }
  __syncthreads();

  // ---- per-thread: gather, loss, rotation trick ----
  const int idxv = s_idx[tid];
  idx_out[(size_t)b * T_ + t0 + tid] = idxv;

  float zt[8], zq[8];
  float lsum = 0.f, nzt2 = 0.f, nzq2 = 0.f;
#pragma unroll
  for (int d = 0; d < 8; ++d) {
    zt[d] = s_zt[tid * 8 + d];
    zq[d] = cb[idxv * 8 + d];
    float df = zt[d] - zq[d];
    lsum = fmaf(df, df, lsum);
    nzt2 = fmaf(zt[d], zt[d], nzt2);
    nzq2 = fmaf(zq[d], zq[d], nzq2);
  }
  const float nzt = sqrtf(nzt2), nzq = sqrtf(nzq2);
  const float ie = s_ie[tid];                       // 1/max(nzt,1e-12)
  const float iq = 1.0f / fmaxf(nzq, 1e-12f);
  float rv[8], en[8], qn[8];
  float rn2 = 0.f;
#pragma unroll
  for (int d = 0; d < 8; ++d) {
    en[d] = zt[d] * ie;
    qn[d] = zq[d] * iq;
    rv[d] = en[d] + qn[d];
    rn2 = fmaf(rv[d], rv[d], rn2);
  }
  const float ir = 1.0f / fmaxf(sqrtf(rn2), 1e-12f);
  float rdot = 0.f, edot = 0.f;
#pragma unroll
  for (int d = 0; d < 8; ++d) {
    rdot = fmaf(rv[d], zt[d], rdot);
    edot = fmaf(en[d], zt[d], edot);
  }
  const float rcoef = 2.0f * ir * ir * rdot;        // 2 * (r.zt) applied to unnorm rv
  const float scal  = nzq / fmaxf(nzt, 1e-8f);
  float* op = &zqrot[((size_t)b * T_ + t0 + tid) * 8];
#pragma unroll
  for (int d = 0; d < 8; ++d)
    op[d] = scal * (zt[d] - rv[d] * rcoef + 2.0f * qn[d] * edot);

  // ---- deterministic loss partial (tree reduce) ----
  s_red[tid] = lsum;
  __syncthreads();
  for (int s = 128; s > 0; s >>= 1) {
    if (tid < s) s_red[tid] += s_red[tid + s];
    __syncthreads();
  }
  if (tid == 0) part[b * NT_TILES + blockIdx.x] = s_red[0];
}

// =========================================================
// Kernel 2: finalize losses (commitment == codebook value)
// =========================================================
__global__ void vq_loss(const float* __restrict__ part,
                        float* __restrict__ commit,
                        float* __restrict__ cbl) {
  int b = threadIdx.x;
  if (b < B_) {
    float s = 0.f;
    for (int i = 0; i < NT_TILES; ++i) s += part[b * NT_TILES + i];
    s *= (1.0f / (float)(T_ * DC));
    commit[b] = s;
    cbl[b]    = s;
  }
}

// =========================================================
// Kernel 3: out_proj via WMMA.  grid (T/16, B), 256 thr
// per wave: B-tile loaded once, 8 o-tiles x 2 chained WMMA
// =========================================================
__global__ __launch_bounds__(256) void vq_outproj(
    const float* __restrict__ out_w,  // [DIN, DC]
    const float* __restrict__ out_b,  // [DIN]
    const float* __restrict__ zqrot,  // [B, T, DC]
    float* __restrict__ zq_out)       // [B, DIN, T]
{
  const int tid   = threadIdx.x;
  const int lane  = tid & 31;
  const int w     = tid >> 5;
  const int r     = lane & 15;
  const int hi    = lane >> 4;
  const int b     = blockIdx.y;
  const int tbase = blockIdx.x * 16;

  // B operand: zq_rot[8 x 16t]; lane n holds col t=tbase+n, K rows 2hi..2hi+1
  const float* br = &zqrot[((size_t)b * T_ + tbase + r) * 8 + hi * 2];
  v2f b0; b0.x = br[0]; b0.y = br[1];
  v2f b1; b1.x = br[4]; b1.y = br[5];

#pragma unroll
  for (int i = 0; i < 8; ++i) {
    const int ot = w * 8 + i;                       // o-tile 0..63
    const float* ar = &out_w[(ot * 16 + r) * 8 + hi * 2];
    v2f a0; a0.x = ar[0]; a0.y = ar[1];
    v2f a1; a1.x = ar[4]; a1.y = ar[5];
    v8f c = {};
    c = __builtin_amdgcn_wmma_f32_16x16x4_f32(false, a0, false, b0,
                                              (short)0, c, false, false);
    c = __builtin_amdgcn_wmma_f32_16x16x4_f32(false, a1, false, b1,
                                              (short)0, c, false, false);
#pragma unroll
    for (int j = 0; j < 8; ++j) {
      const int o = ot * 16 + hi * 8 + j;           // D: vgpr j -> rows j, j+8
      zq_out[((size_t)b * DIN + o) * T_ + tbase + r] = c[j] + out_b[o];
    }
  }
}

// =========================================================
extern "C" void kernel_launch(void* const* d_in, const int* in_sizes, int n_in,
                              void* d_out, int out_size, void* d_ws, size_t ws_size,
                              hipStream_t stream) {
  const float* z     = (const float*)d_in[0];
  const float* in_w  = (const float*)d_in[1];
  const float* in_b  = (const float*)d_in[2];
  const float* out_w = (const float*)d_in[3];
  const float* out_b = (const float*)d_in[4];
  const float* cb    = (const float*)d_in[5];

  float* out    = (float*)d_out;
  float* zq_out = out + OUT_ZQ;
  float* commit = out + OUT_COMMIT;
  float* cbl    = out + OUT_CBL;
  int*   idxo   = (int*)(out + OUT_IDX);
  float* ze     = out + OUT_ZE;

  float* ws    = (float*)d_ws;
  float* cn    = ws + WS_CN;
  float* cn2   = ws + WS_CN2;
  float* zqr   = ws + WS_ZQROT;
  float* part  = ws + WS_PART;

  vq_prep<<<dim3((K_ + 255) / 256), dim3(256), 0, stream>>>(cb, cn, cn2);
  vq_main<<<dim3(NT_TILES, B_), dim3(256), 0, stream>>>(
      z, in_w, in_b, cb, cn, cn2, ze, idxo, zqr, part);
  vq_loss<<<dim3(1), dim3(32), 0, stream>>>(part, commit, cbl);
  vq_outproj<<<dim3(T_ / 16, B_), dim3(256), 0, stream>>>(out_w, out_b, zqr, zq_out);
}